// Model_25795573580198
// MI455X (gfx1250) — compile-verified
//
#include <hip/hip_runtime.h>

typedef float v2f __attribute__((ext_vector_type(2)));
typedef float v8f __attribute__((ext_vector_type(8)));

#define N_NODES   100000
#define D_FEAT    128
#define N_HID     128
#define N_CLASSES 40

// ---------------------------------------------------------------- degree prep
__global__ __launch_bounds__(256) void init_deg_kernel(float* deg, int n) {
    int i = blockIdx.x * 256 + threadIdx.x;
    if (i < n) deg[i] = 1.0f;  // self-loop contributes 1 to in-degree
}

__global__ __launch_bounds__(256) void deg_accum_kernel(const long long* __restrict__ dst,
                                                        float* deg, int e) {
    int i = blockIdx.x * 256 + threadIdx.x;
    if (i < e) {
        int d = (int)dst[i];
        __hip_atomic_fetch_add(&deg[d], 1.0f, __ATOMIC_RELAXED, __HIP_MEMORY_SCOPE_AGENT);
    }
}

__global__ __launch_bounds__(256) void dinv_kernel(float* deg, int n) {
    int i = blockIdx.x * 256 + threadIdx.x;
    if (i < n) {
        float d = deg[i];
        deg[i] = (d > 0.0f) ? rsqrtf(d) : 0.0f;  // v_rsq_f32
    }
}

// ------------------------------------------------- GEMM1: h = x @ W1 (f32 WMMA)
// Block: 16 rows of x; 8 waves, wave w -> output col tile [16w, 16w+16).
// A 16x4 f32 layout: lanes 0-15 M=lane K={0,1}; lanes 16-31 M=lane-16 K={2,3}.
// B 4x16 f32 layout: VGPR v: lanes 0-15 K=v N=lane; lanes 16-31 K=v+2 N=lane-16.
// C/D 16x16 f32: VGPR r: lanes 0-15 (M=r,N=lane); lanes 16-31 (M=r+8,N=lane-16).
__global__ __launch_bounds__(256) void gemm1_wmma(const float* __restrict__ x,
                                                  const float* __restrict__ W1,
                                                  float* __restrict__ h) {
    const int lane = threadIdx.x & 31;
    const int wave = threadIdx.x >> 5;
    const int r0   = blockIdx.x * 16;
    const int n0   = wave * 16;
    const int m    = lane & 15;
    const int half = lane >> 4;

    const float* __restrict__ xrow = x + (size_t)(r0 + m) * D_FEAT;
    v8f c = {};
#pragma unroll 4
    for (int k0 = 0; k0 < D_FEAT; k0 += 4) {
        const int kA = k0 + 2 * half;
        v2f a, b;
        a.x = xrow[kA];
        a.y = xrow[kA + 1];
        b.x = W1[(size_t)kA * N_HID + n0 + m];
        b.y = W1[(size_t)(kA + 1) * N_HID + n0 + m];
        c = __builtin_amdgcn_wmma_f32_16x16x4_f32(false, a, false, b,
                                                  (short)0, c, false, false);
    }
#pragma unroll
    for (int r = 0; r < 8; ++r) {
        const int row = r0 + r + 8 * half;
        h[(size_t)row * N_HID + n0 + m] = c[r];
    }
}

// ------------------------------------------- agg init: self-loop term h*dinv^2
__global__ __launch_bounds__(256) void selfloop_kernel(const float* __restrict__ h,
                                                       const float* __restrict__ dinv,
                                                       float* __restrict__ agg, int total) {
    int idx = blockIdx.x * 256 + threadIdx.x;
    if (idx < total) {
        int node = idx >> 7;  // /128
        float di = dinv[node];
        agg[idx] = h[idx] * (di * di);
    }
}

// ---------------------------------- edge aggregation: one wave32 per edge
// lane handles 4 features (float4 gather, 4 non-returning f32 atomics).
__global__ __launch_bounds__(256) void edge_agg_kernel(const long long* __restrict__ src,
                                                       const long long* __restrict__ dst,
                                                       const float* __restrict__ dinv,
                                                       const float* __restrict__ h,
                                                       float* __restrict__ agg, int e) {
    const int lane = threadIdx.x & 31;
    const int wid  = blockIdx.x * 8 + (threadIdx.x >> 5);
    if (wid >= e) return;  // wave-uniform branch

    const int s = (int)src[wid];
    const int d = (int)dst[wid];
    const float norm = dinv[s] * dinv[d];

    const float4* __restrict__ hp = (const float4*)(h + (size_t)s * N_HID);
    float4 v = hp[lane];
    float* ap = agg + (size_t)d * N_HID + lane * 4;
    __hip_atomic_fetch_add(ap + 0, v.x * norm, __ATOMIC_RELAXED, __HIP_MEMORY_SCOPE_AGENT);
    __hip_atomic_fetch_add(ap + 1, v.y * norm, __ATOMIC_RELAXED, __HIP_MEMORY_SCOPE_AGENT);
    __hip_atomic_fetch_add(ap + 2, v.z * norm, __ATOMIC_RELAXED, __HIP_MEMORY_SCOPE_AGENT);
    __hip_atomic_fetch_add(ap + 3, v.w * norm, __ATOMIC_RELAXED, __HIP_MEMORY_SCOPE_AGENT);
}

// ---------------- GEMM2: logits = relu(agg + b1) @ Wc + bc  (f32 WMMA, N=40->48)
__global__ __launch_bounds__(96) void gemm2_wmma(const float* __restrict__ agg,
                                                 const float* __restrict__ b1,
                                                 const float* __restrict__ Wc,
                                                 const float* __restrict__ bc,
                                                 float* __restrict__ out) {
    const int lane = threadIdx.x & 31;
    const int wave = threadIdx.x >> 5;  // 0..2 -> col tiles 0,16,32
    const int r0   = blockIdx.x * 16;
    const int n0   = wave * 16;
    const int m    = lane & 15;
    const int half = lane >> 4;
    const int n    = n0 + m;
    const bool nok = (n < N_CLASSES);

    const float* __restrict__ arow = agg + (size_t)(r0 + m) * N_HID;
    v8f c = {};
#pragma unroll 4
    for (int k0 = 0; k0 < N_HID; k0 += 4) {
        const int kA = k0 + 2 * half;
        v2f a, b;
        a.x = fmaxf(arow[kA] + b1[kA], 0.0f);        // fused bias + ReLU
        a.y = fmaxf(arow[kA + 1] + b1[kA + 1], 0.0f);
        b.x = nok ? Wc[(size_t)kA * N_CLASSES + n] : 0.0f;
        b.y = nok ? Wc[(size_t)(kA + 1) * N_CLASSES + n] : 0.0f;
        c = __builtin_amdgcn_wmma_f32_16x16x4_f32(false, a, false, b,
                                                  (short)0, c, false, false);
    }
    if (nok) {
        const float bcv = bc[n];
#pragma unroll
        for (int r = 0; r < 8; ++r) {
            const int row = r0 + r + 8 * half;
            out[(size_t)row * N_CLASSES + n] = c[r] + bcv;
        }
    }
}

// --------------------------------------------------------------------- launch
extern "C" void kernel_launch(void* const* d_in, const int* in_sizes, int n_in,
                              void* d_out, int out_size, void* d_ws, size_t ws_size,
                              hipStream_t stream) {
    const float*     x   = (const float*)d_in[0];
    const long long* ei  = (const long long*)d_in[1];  // int64 [2, E]
    // d_in[2] = conv_time (unused by the reference computation)
    const float*     W1  = (const float*)d_in[3];
    const float*     b1  = (const float*)d_in[4];
    const float*     Wc  = (const float*)d_in[5];
    const float*     bc  = (const float*)d_in[6];
    float*           out = (float*)d_out;

    const int E = in_sizes[1] / 2;
    const long long* src = ei;
    const long long* dst = ei + E;

    // Workspace layout (floats): dinv[100352] | h[100000*128] | agg[100000*128]
    float* ws   = (float*)d_ws;
    float* dinv = ws;                                  // holds deg, then rsqrt in place
    float* h    = ws + 100352;                         // 128-aligned
    float* agg  = h + (size_t)N_NODES * N_HID;

    const int total = N_NODES * N_HID;  // 12.8M, fits int

    init_deg_kernel <<<(N_NODES + 255) / 256, 256, 0, stream>>>(dinv, N_NODES);
    deg_accum_kernel<<<(E + 255) / 256,      256, 0, stream>>>(dst, dinv, E);
    dinv_kernel     <<<(N_NODES + 255) / 256, 256, 0, stream>>>(dinv, N_NODES);

    gemm1_wmma      <<<N_NODES / 16, 256, 0, stream>>>(x, W1, h);

    selfloop_kernel <<<(total + 255) / 256, 256, 0, stream>>>(h, dinv, agg, total);
    edge_agg_kernel <<<(E + 7) / 8, 256, 0, stream>>>(src, dst, dinv, h, agg, E);

    gemm2_wmma      <<<N_NODES / 16, 96, 0, stream>>>(agg, b1, Wc, bc, out);
}